// SemLA_50285477102153
// MI455X (gfx1250) — compile-verified
//
#include <hip/hip_runtime.h>

typedef __attribute__((ext_vector_type(2))) float v2f;
typedef __attribute__((ext_vector_type(8))) float v8f;

constexpr int   Cc    = 64;
constexpr int   Hh    = 80;
constexpr int   Ww    = 80;
constexpr int   L     = Hh * Ww;        // 6400
constexpr int   TILES = L / 16;         // 400
constexpr float NEGV  = -1e9f;

// output layout (floats, reference return order)
constexpr int OFF_MK0   = 0;            // [6400,2]
constexpr int OFF_MK1   = 12800;        // [1,6400,2]
constexpr int OFF_MASKV = 25600;        // [1,6400]
constexpr int OFF_SCORE = 32000;        // [1,6400]
constexpr int OFF_UP    = 38400;        // [1,1,640,640]

// ---------------------------------------------------------------------------
// 16x16 output tile of  (A[:,i0:i0+16])^T * (B[:,j0:j0+16]),  K = 64,
// via 16 chained V_WMMA_F32_16X16X4_F32.  Inputs pre-scaled by 1/sqrt(C)=0.125.
// A operand regs are preloaded per wave (a[16] of v2f); B is loaded per tile.
// ISA layout: lanes 0-15 hold K=+0 (reg .x) / K=+1 (.y); lanes 16-31 K=+2/+3.
// ---------------------------------------------------------------------------
__device__ __forceinline__ v8f tile_gemm(const float* __restrict__ B,
                                         int j0, int half, int m,
                                         const v2f (&a)[16]) {
    v8f acc = {};
#pragma unroll
    for (int kk = 0; kk < 16; ++kk) {
        const int c = 4 * kk + 2 * half;
        v2f b;
        b.x = B[(c + 0) * L + j0 + m] * 0.125f;
        b.y = B[(c + 1) * L + j0 + m] * 0.125f;
        acc = __builtin_amdgcn_wmma_f32_16x16x4_f32(
            /*neg_a=*/false, a[kk], /*neg_b=*/false, b,
            /*c_mod=*/(short)0, acc, /*reuse_a=*/false, /*reuse_b=*/false);
    }
    return acc;
}

__device__ __forceinline__ void load_a(const float* __restrict__ A,
                                       int i0, int half, int m, v2f (&a)[16]) {
#pragma unroll
    for (int kk = 0; kk < 16; ++kk) {
        const int c = 4 * kk + 2 * half;
        a[kk].x = A[(c + 0) * L + i0 + m] * 0.125f;
        a[kk].y = A[(c + 1) * L + i0 + m] * 0.125f;
    }
}

// ---------------------------------------------------------------------------
// Pass 1: row_max (mode 0: A=vi,B=ir) and col_max (mode 1: A=ir,B=vi).
// One wave per 16-row strip; loops over all 400 column tiles.
// ---------------------------------------------------------------------------
__global__ void __launch_bounds__(256)
semla_maxpass(const float* __restrict__ vi, const float* __restrict__ ir,
              const float* __restrict__ sa_vi, const float* __restrict__ sa_ir,
              float* __restrict__ row_max, float* __restrict__ col_max) {
    const int wave = threadIdx.x >> 5;
    const int lane = threadIdx.x & 31;
    const int half = lane >> 4;
    const int m    = lane & 15;

    const int mode = (blockIdx.x >= 50) ? 1 : 0;
    const int tile = (blockIdx.x - 50 * mode) * 8 + wave;   // 0..399
    const int i0   = tile * 16;

    const float* A    = mode ? ir    : vi;
    const float* B    = mode ? vi    : ir;
    const float* saA  = mode ? sa_ir : sa_vi;
    const float* saB  = mode ? sa_vi : sa_ir;
    float* outmax     = mode ? col_max : row_max;

    v2f a[16];
    load_a(A, i0, half, m, a);

    bool rvalid[8];
#pragma unroll
    for (int v = 0; v < 8; ++v) rvalid[v] = saA[i0 + v + 8 * half] > 0.0f;

    float rmax[8];
#pragma unroll
    for (int v = 0; v < 8; ++v) rmax[v] = NEGV;

    for (int jt = 0; jt < TILES; ++jt) {
        const int  j0     = jt * 16;
        const bool cvalid = saB[j0 + m] > 0.0f;
        v8f d = tile_gemm(B, j0, half, m, a);
#pragma unroll
        for (int v = 0; v < 8; ++v) {
            const float val = (rvalid[v] && cvalid) ? d[v] * 10.0f : NEGV;
            rmax[v] = fmaxf(rmax[v], val);
        }
    }

    // max over the 16 lanes sharing each row (xor masks <16 stay in-half)
#pragma unroll
    for (int v = 0; v < 8; ++v) {
#pragma unroll
        for (int off = 8; off >= 1; off >>= 1)
            rmax[v] = fmaxf(rmax[v], __shfl_xor(rmax[v], off, 32));
    }
    if (m == 0) {
#pragma unroll
        for (int v = 0; v < 8; ++v) outmax[i0 + v + 8 * half] = rmax[v];
    }
}

// ---------------------------------------------------------------------------
// Pass 2: recompute tiles (bit-identical to pass-1 mode 0) and find, per row,
// the smallest j with  val==row_max[i] && val==col_max[j] && val>0.
// Emits mkpts1 / mask_v / score.
// ---------------------------------------------------------------------------
__global__ void __launch_bounds__(256)
semla_matchpass(const float* __restrict__ vi, const float* __restrict__ ir,
                const float* __restrict__ sa_vi, const float* __restrict__ sa_ir,
                const float* __restrict__ row_max, const float* __restrict__ col_max,
                float* __restrict__ out) {
    const int wave = threadIdx.x >> 5;
    const int lane = threadIdx.x & 31;
    const int half = lane >> 4;
    const int m    = lane & 15;

    const int tile = blockIdx.x * 8 + wave;   // 0..399
    const int i0   = tile * 16;

    v2f a[16];
    load_a(vi, i0, half, m, a);

    bool  rvalid[8];
    float rmaxv[8];
#pragma unroll
    for (int v = 0; v < 8; ++v) {
        const int row = i0 + v + 8 * half;
        rvalid[v] = sa_vi[row] > 0.0f;
        rmaxv[v]  = row_max[row];
    }

    int   bestj[8];
    float bests[8];
#pragma unroll
    for (int v = 0; v < 8; ++v) { bestj[v] = 0x3FFFFFFF; bests[v] = 0.0f; }

    for (int jt = 0; jt < TILES; ++jt) {
        const int   j0     = jt * 16;
        const bool  cvalid = sa_ir[j0 + m] > 0.0f;
        const float cmaxv  = col_max[j0 + m];
        v8f d = tile_gemm(ir, j0, half, m, a);
#pragma unroll
        for (int v = 0; v < 8; ++v) {
            const float val  = (rvalid[v] && cvalid) ? d[v] * 10.0f : NEGV;
            const bool  cond = (val == rmaxv[v]) && (val == cmaxv) && (val > 0.0f);
            const int   j    = j0 + m;
            if (cond && j < bestj[v]) { bestj[v] = j; bests[v] = val; }
        }
    }

    // min-j (with payload) over the 16 lanes sharing each row
#pragma unroll
    for (int v = 0; v < 8; ++v) {
#pragma unroll
        for (int off = 8; off >= 1; off >>= 1) {
            const int   oj = __shfl_xor(bestj[v], off, 32);
            const float os = __shfl_xor(bests[v], off, 32);
            if (oj < bestj[v]) { bestj[v] = oj; bests[v] = os; }
        }
    }

    if (m == 0) {
#pragma unroll
        for (int v = 0; v < 8; ++v) {
            const int  row   = i0 + v + 8 * half;
            const bool found = bestj[v] < 0x3FFFFFFF;
            const int  aj    = found ? bestj[v] : 0;        // argmax(all-zero)==0
            out[OFF_MK1 + 2 * row + 0] = (float)((aj % Ww) * 8);
            out[OFF_MK1 + 2 * row + 1] = (float)((aj / Ww) * 8);
            out[OFF_MASKV + row]       = found ? 1.0f : 0.0f;
            out[OFF_SCORE + row]       = found ? bests[v] : 0.0f;   // conf[i,j*]*mask
        }
    }
}

// ---------------------------------------------------------------------------
// mkpts0 (static grid of keypoints)
// ---------------------------------------------------------------------------
__global__ void semla_mkpts0(float* __restrict__ out) {
    const int l = blockIdx.x * blockDim.x + threadIdx.x;
    if (l < L) {
        out[OFF_MK0 + 2 * l + 0] = (float)((l % Ww) * 8);
        out[OFF_MK0 + 2 * l + 1] = (float)((l / Ww) * 8);
    }
}

// ---------------------------------------------------------------------------
// Bilinear upsample x8, align_corners=True (80x80 -> 640x640)
// ---------------------------------------------------------------------------
__global__ void semla_upsample(const float* __restrict__ sa, float* __restrict__ out) {
    const int o = blockIdx.x * blockDim.x + threadIdx.x;
    if (o >= 640 * 640) return;
    const int oy = o / 640, ox = o % 640;
    const float step = 79.0f / 639.0f;
    const float fy = oy * step;
    const float fx = ox * step;
    int y0 = (int)floorf(fy); int y1 = min(y0 + 1, Hh - 1);
    int x0 = (int)floorf(fx); int x1 = min(x0 + 1, Ww - 1);
    const float wy = fy - (float)y0;
    const float wx = fx - (float)x0;
    // match reference order: interpolate H first, then W
    const float r0 = sa[y0 * Ww + x0] * (1.0f - wy) + sa[y1 * Ww + x0] * wy;
    const float r1 = sa[y0 * Ww + x1] * (1.0f - wy) + sa[y1 * Ww + x1] * wy;
    out[o] = r0 * (1.0f - wx) + r1 * wx;
}

// ---------------------------------------------------------------------------
extern "C" void kernel_launch(void* const* d_in, const int* in_sizes, int n_in,
                              void* d_out, int out_size, void* d_ws, size_t ws_size,
                              hipStream_t stream) {
    const float* vi    = (const float*)d_in[0];   // [1,64,80,80]
    const float* ir    = (const float*)d_in[1];   // [1,64,80,80]
    const float* sa_vi = (const float*)d_in[2];   // [1,1,80,80]
    const float* sa_ir = (const float*)d_in[3];   // [1,1,80,80]
    float* out = (float*)d_out;

    float* row_max = (float*)d_ws;        // L floats
    float* col_max = row_max + L;         // L floats

    // pass 1: 50 blocks row-max + 50 blocks col-max, 8 waves (tiles) per block
    semla_maxpass<<<100, 256, 0, stream>>>(vi, ir, sa_vi, sa_ir, row_max, col_max);
    // pass 2: mutual-NN match
    semla_matchpass<<<50, 256, 0, stream>>>(vi, ir, sa_vi, sa_ir, row_max, col_max, out);
    // static keypoint grid
    semla_mkpts0<<<(L + 255) / 256, 256, 0, stream>>>(out);
    // saliency upsample
    semla_upsample<<<(640 * 640 + 255) / 256, 256, 0, stream>>>(sa_ir, out + OFF_UP);
}